// MobiusDist2Hyperplane_16595753632517
// MI455X (gfx1250) — compile-verified
//
#include <hip/hip_runtime.h>
#include <hip/hip_bf16.h>
#include <math.h>

// Mobius dist2plane, fused dual-GEMM (x@p.T and x@a.T) + analytic epilogue.
// Matrix cores: V_WMMA_F32_16X16X4_F32 (fp32 in, fp32 accum).
// Data movement: GLOBAL_LOAD_ASYNC_TO_LDS_B128 (ASYNCcnt) direct-to-LDS staging.

typedef __attribute__((ext_vector_type(2))) float v2f;
typedef __attribute__((ext_vector_type(8))) float v8f;

#define D_DIM 512
#define O_DIM 512
#define TILE_N 128
#define TILE_O 64
#define KC 16
#define LDSTR 20   // padded LDS row stride (floats): conflict-free, float4-aligned
#define MINN 1e-15f

// ---- async 16B global -> LDS copy (per-lane), tracked by ASYNCcnt ---------
__device__ __forceinline__ void async_b128(const float* g, const float* l) {
  unsigned lds = (unsigned)(size_t)l;   // low 32 bits of generic ptr = LDS offset
  asm volatile("global_load_async_to_lds_b128 %0, %1, off"
               :: "v"(lds), "v"(g) : "memory");
}
__device__ __forceinline__ void wait_async0() {
  asm volatile("s_wait_asynccnt 0x0" ::: "memory");
}

// ---- per-row reductions: x2[n] = ||x_n||^2 --------------------------------
__global__ __launch_bounds__(256) void mobius_precompute_x2(
    const float* __restrict__ x, float* __restrict__ x2, int N) {
  int n = blockIdx.x * 8 + (threadIdx.x >> 5);
  int lane = threadIdx.x & 31;
  if (n >= N) return;
  float s = 0.f;
  for (int k = lane; k < D_DIM; k += 32) {
    float v = x[(size_t)n * D_DIM + k];
    s = fmaf(v, v, s);
  }
#pragma unroll
  for (int off = 16; off; off >>= 1) s += __shfl_down(s, off, 32);
  if (lane == 0) x2[n] = s;
}

// ---- per-o reductions: p2, <p,a>, ||a||, exp(scale) -----------------------
__global__ __launch_bounds__(256) void mobius_precompute_o(
    const float* __restrict__ p, const float* __restrict__ a,
    const float* __restrict__ scale,
    float* __restrict__ p2v, float* __restrict__ pav,
    float* __restrict__ anv, float* __restrict__ esv, int O) {
  int o = blockIdx.x * 8 + (threadIdx.x >> 5);
  int lane = threadIdx.x & 31;
  if (o >= O) return;
  float sp2 = 0.f, spa = 0.f, sa2 = 0.f;
  for (int k = lane; k < D_DIM; k += 32) {
    float pv = p[(size_t)o * D_DIM + k];
    float av = a[(size_t)o * D_DIM + k];
    sp2 = fmaf(pv, pv, sp2);
    spa = fmaf(pv, av, spa);
    sa2 = fmaf(av, av, sa2);
  }
#pragma unroll
  for (int off = 16; off; off >>= 1) {
    sp2 += __shfl_down(sp2, off, 32);
    spa += __shfl_down(spa, off, 32);
    sa2 += __shfl_down(sa2, off, 32);
  }
  if (lane == 0) {
    p2v[o] = sp2;
    pav[o] = spa;
    anv[o] = fmaxf(sqrtf(sa2), MINN);
    esv[o] = expf(scale[o]);
  }
}

// ---- main fused kernel ----------------------------------------------------
__global__ __launch_bounds__(256) void mobius_dist2plane_wmma(
    const float* __restrict__ x, const float* __restrict__ p,
    const float* __restrict__ a,
    const float* __restrict__ x2v, const float* __restrict__ p2v,
    const float* __restrict__ pav, const float* __restrict__ anv,
    const float* __restrict__ esv, float* __restrict__ out) {
  __shared__ float sx[2][TILE_N * LDSTR];   // x tile   128 x 16
  __shared__ float sp[2][TILE_O * LDSTR];   // p tile    64 x 16
  __shared__ float sa[2][TILE_O * LDSTR];   // a tile    64 x 16

  const int tid  = threadIdx.x;
  const int lane = tid & 31;
  const int wave = tid >> 5;     // 0..7
  const int wn = wave & 3;       // wave's N quadrant (0..3) -> 32 rows each
  const int wo = wave >> 2;      // wave's O half     (0..1) -> 32 cols each
  const int nBlock = blockIdx.y * TILE_N;
  const int oBlock = blockIdx.x * TILE_O;

  // fragment addressing (A 16x4 and B 4x16 share the same pattern):
  const int fr = lane & 15;          // row within 16
  const int fk = (lane >> 4) << 1;   // k sub-offset {0, 2}

  v8f accP[2][2], accA[2][2];
#pragma unroll
  for (int i = 0; i < 2; ++i)
#pragma unroll
    for (int j = 0; j < 2; ++j) {
      accP[i][j] = (v8f)(0.0f);
      accA[i][j] = (v8f)(0.0f);
    }

  // Issue one K-chunk of async global->LDS copies (4 x b128 per thread).
  auto issueChunk = [&](int buf, int kb) {
    // x: 128 rows x 16 cols = 512 x b128, 2 per thread
#pragma unroll
    for (int i = 0; i < 2; ++i) {
      int q = tid + i * 256;
      int row = q >> 2;
      int c4  = (q & 3) << 2;
      async_b128(x + (size_t)(nBlock + row) * D_DIM + kb + c4,
                 &sx[buf][row * LDSTR + c4]);
    }
    // p, a: 64 rows x 16 cols = 256 x b128 each, 1 per thread
    {
      int row = tid >> 2;
      int c4  = (tid & 3) << 2;
      async_b128(p + (size_t)(oBlock + row) * D_DIM + kb + c4,
                 &sp[buf][row * LDSTR + c4]);
      async_b128(a + (size_t)(oBlock + row) * D_DIM + kb + c4,
                 &sa[buf][row * LDSTR + c4]);
    }
  };

  issueChunk(0, 0);
  const int nChunks = D_DIM / KC;  // 32
  for (int c = 0; c < nChunks; ++c) {
    wait_async0();        // this wave's chunk-c copies have landed in LDS
    __syncthreads();      // all waves landed; all done reading buf being reused
    if (c + 1 < nChunks) issueChunk((c + 1) & 1, (c + 1) * KC);

    const int buf = c & 1;
    const float* xs  = &sx[buf][0];
    const float* ps  = &sp[buf][0];
    const float* as_ = &sa[buf][0];
#pragma unroll
    for (int k0 = 0; k0 < KC; k0 += 4) {
      v2f af[2], bp[2], ba[2];
#pragma unroll
      for (int mt = 0; mt < 2; ++mt)
        af[mt] = *(const v2f*)(xs + (wn * 32 + mt * 16 + fr) * LDSTR + k0 + fk);
#pragma unroll
      for (int ot = 0; ot < 2; ++ot) {
        bp[ot] = *(const v2f*)(ps  + (wo * 32 + ot * 16 + fr) * LDSTR + k0 + fk);
        ba[ot] = *(const v2f*)(as_ + (wo * 32 + ot * 16 + fr) * LDSTR + k0 + fk);
      }
#pragma unroll
      for (int mt = 0; mt < 2; ++mt)
#pragma unroll
        for (int ot = 0; ot < 2; ++ot) {
          accP[mt][ot] = __builtin_amdgcn_wmma_f32_16x16x4_f32(
              false, af[mt], false, bp[ot], (short)0, accP[mt][ot], false, false);
          accA[mt][ot] = __builtin_amdgcn_wmma_f32_16x16x4_f32(
              false, af[mt], false, ba[ot], (short)0, accA[mt][ot], false, false);
        }
    }
  }

  // ---- fused Mobius epilogue on accumulators ------------------------------
  // C/D layout: VGPR r, lanes 0-15 -> M=r, lanes 16-31 -> M=r+8; N = lane&15
  const int rHi = (lane >> 4) << 3;
#pragma unroll
  for (int mt = 0; mt < 2; ++mt) {
#pragma unroll
    for (int ot = 0; ot < 2; ++ot) {
      const int o  = oBlock + wo * 32 + ot * 16 + fr;
      const float p2 = p2v[o];
      const float pa = pav[o];
      const float an = anv[o];
      const float es = esv[o];
      const float Bc = 1.f - p2;
      const int n0 = nBlock + wn * 32 + mt * 16 + rHi;
#pragma unroll
      for (int r = 0; r < 8; ++r) {
        const int n = n0 + r;
        const float x2 = x2v[n];
        const float xp = accP[mt][ot][r];
        const float xa = accA[mt][ot][r];
        const float Ac = 1.f - 2.f * xp + x2;                       // c == 1
        const float Dn = fmaxf(1.f - 2.f * xp + p2 * x2, MINN);
        float dn2 = (Ac * Ac * p2 - 2.f * Ac * Bc * xp + Bc * Bc * x2) / (Dn * Dn);
        dn2 = fmaxf(dn2, MINN);
        const float sda   = (Bc * xa - Ac * pa) / Dn;
        const float denom = fmaxf((1.f - dn2) * an, MINN);
        out[(size_t)n * O_DIM + o] = asinhf((2.f * sda) / denom) * es;
      }
    }
  }
}

extern "C" void kernel_launch(void* const* d_in, const int* in_sizes, int n_in,
                              void* d_out, int out_size, void* d_ws, size_t ws_size,
                              hipStream_t stream) {
  (void)n_in; (void)out_size; (void)ws_size;
  const float* x     = (const float*)d_in[0];
  const float* p     = (const float*)d_in[1];
  const float* a     = (const float*)d_in[2];
  const float* scale = (const float*)d_in[3];
  const int N = in_sizes[0] / D_DIM;   // 16384
  const int O = in_sizes[3];           // 512

  float* ws  = (float*)d_ws;
  float* x2v = ws;
  float* p2v = x2v + N;
  float* pav = p2v + O;
  float* anv = pav + O;
  float* esv = anv + O;
  float* out = (float*)d_out;

  mobius_precompute_x2<<<dim3((N + 7) / 8), dim3(256), 0, stream>>>(x, x2v, N);
  mobius_precompute_o<<<dim3((O + 7) / 8), dim3(256), 0, stream>>>(
      p, a, scale, p2v, pav, anv, esv, O);
  mobius_dist2plane_wmma<<<dim3(O / TILE_O, N / TILE_N), dim3(256), 0, stream>>>(
      x, p, a, x2v, p2v, pav, anv, esv, out);
}